// PCEN_23038204575927
// MI455X (gfx1250) — compile-verified
//
#include <hip/hip_runtime.h>
#include <hip/hip_bf16.h>

typedef __attribute__((ext_vector_type(2))) float v2f;
typedef __attribute__((ext_vector_type(8))) float v8f;

#define SM      0.04f     // EMA smoothing coefficient s
#define OMS     0.96f     // 1 - s
#define PFLOOR  1e-6f
#define SEGLEN  1024      // T-segment length (T=4096 -> 4 segments)
#define WARM    512       // redundant warm-up steps; 0.96^512 ~ 8e-10 << f32 eps
#define PF_CHUNKS 8       // prefetch distance in 16-step chunks (8 KB/wave ahead)

// One wave32 owns a 16-channel slice and scans T sequentially in 16-step
// chunks. Each chunk: H[16x16] = L[16x16] * X[16x16] + d (x) h_prev, computed
// as 4 chained V_WMMA_F32_16X16X4_F32 (K slices 0..3 / 4..7 / 8..11 / 12..15).
__global__ __launch_bounds__(256) void pcen_wmma_kernel(
    const float* __restrict__ x, const float* __restrict__ alpha,
    const float* __restrict__ delta, const float* __restrict__ root,
    float* __restrict__ out, int B, int T, int C)
{
  const int lane = threadIdx.x & 31;
  const int wv   = threadIdx.x >> 5;   // wave id in block: channel group
  const int hi   = lane >> 4;          // half-wave select (lanes 16-31)
  const int n    = lane & 15;          // channel within group / N index
  const int b    = blockIdx.y;
  const int seg  = blockIdx.x;
  const int c0   = wv << 4;
  const int c    = c0 + n;

  // ---- per-channel parameters (hoisted out of the scan) ----
  const float a_c = fminf(alpha[c], 1.0f);
  const float oor = 1.0f / fmaxf(root[c], 1.0f);
  const float dlt = delta[c];
  const float dk  = __builtin_amdgcn_exp2f(oor * __builtin_amdgcn_logf(dlt)); // delta^(1/root)

  // ---- constant A matrix (L, lower triangular EMA kernel), WMMA A layout:
  //      16x4 f32 slice q: VGPR0 -> K=4q+2*hi, VGPR1 -> K=4q+1+2*hi; M = lane&15
  v2f Amat[4];
  #pragma unroll
  for (int q = 0; q < 4; ++q) {
    #pragma unroll
    for (int e = 0; e < 2; ++e) {
      const int K = 4 * q + e + 2 * hi;
      Amat[q][e] = (K <= n) ? SM * __powf(OMS, (float)(n - K)) : 0.0f;
    }
  }
  // carry decay vector in C/D layout: row M = r + 8*hi gets (1-s)^(M+1)
  float dvec[8];
  #pragma unroll
  for (int r = 0; r < 8; ++r)
    dvec[r] = __powf(OMS, (float)(r + 8 * hi + 1));

  const int segStart  = seg * SEGLEN;
  const int warmStart = (segStart >= WARM) ? (segStart - WARM) : 0;
  const int nChunks   = (segStart + SEGLEN - warmStart) >> 4;
  const int outChunk0 = (segStart - warmStart) >> 4;   // warm-up chunks emit nothing

  const float* xb = x   + (size_t)b * T * C;
  float*       ob = out + (size_t)b * T * C;

  // h_{-1} = x at the segment's warm start (exact for seg 0; decayed away else)
  float carry = xb[(size_t)warmStart * C + c];

  __shared__ float lds[8 * 256];          // 1 KB tile per wave, wave-private
  float* tile = &lds[wv * 256];

  for (int i = 0; i < nChunks; ++i) {
    const int t0 = warmStart + (i << 4);
    const float* src = xb + (size_t)t0 * C + c0;

    // ---- deep software prefetch: global_prefetch_b8, no LOADcnt pressure ----
    {
      const int tp = t0 + (PF_CHUNKS << 4);
      if (tp + 16 <= segStart + SEGLEN) {   // uniform across wave
        const char* p = (const char*)(xb + (size_t)(tp + n) * C + c0) + (hi << 5);
        __builtin_prefetch(p, 0, 1);
      }
    }

    // ---- load X chunk directly in WMMA B layout (4x16 f32 slices) ----
    v2f Bm[4];
    #pragma unroll
    for (int q = 0; q < 4; ++q) {
      const int tA = 4 * q + 2 * hi;
      Bm[q][0] = src[(size_t)tA      * C + n];
      Bm[q][1] = src[(size_t)(tA + 1) * C + n];
    }

    // ---- C = d (outer) h_prev, then H = L*X + C via 4 chained f32 WMMAs ----
    v8f acc;
    #pragma unroll
    for (int r = 0; r < 8; ++r) acc[r] = dvec[r] * carry;

    #pragma unroll
    for (int q = 0; q < 4; ++q)
      acc = __builtin_amdgcn_wmma_f32_16x16x4_f32(false, Amat[q], false, Bm[q],
                                                  (short)0, acc, false, false);

    // next carry = H[15][n] : C/D VGPR 7, lanes 16..31 hold M=15
    carry = __shfl(acc[7], 16 + n, 32);

    if (i >= outChunk0) {
      // stage x through wave-private LDS: B layout -> C/D layout
      #pragma unroll
      for (int q = 0; q < 4; ++q) {
        const int tA = 4 * q + 2 * hi;
        tile[tA       * 16 + n] = Bm[q][0];
        tile[(tA + 1) * 16 + n] = Bm[q][1];
      }
      asm volatile("" ::: "memory");  // DS ops are in-order within a wave

      float* dst = ob + (size_t)t0 * C + c0;
      #pragma unroll
      for (int r = 0; r < 8; ++r) {
        const int t  = r + (hi << 3);            // C/D row M
        const float xv = tile[t * 16 + n];
        const float h  = acc[r];
        // (FLOOR+h)^-alpha on the trans pipe: v_log_f32 + v_exp_f32; fold the
        // division into an fma (x * (FLOOR+ema)^-a + delta)
        const float pinv = __builtin_amdgcn_exp2f(-a_c * __builtin_amdgcn_logf(PFLOOR + h));
        const float y    = fmaf(xv, pinv, dlt);
        const float z    = __builtin_amdgcn_exp2f(oor * __builtin_amdgcn_logf(y));
        dst[(size_t)t * C + n] = z - dk;
      }
    }
  }
}

extern "C" void kernel_launch(void* const* d_in, const int* in_sizes, int n_in,
                              void* d_out, int out_size, void* d_ws, size_t ws_size,
                              hipStream_t stream) {
  const float* x     = (const float*)d_in[0];
  const float* alpha = (const float*)d_in[1];
  const float* delta = (const float*)d_in[2];
  const float* root  = (const float*)d_in[3];
  float* out = (float*)d_out;

  const int C = in_sizes[1];             // 128
  const int T = 4096;
  const int B = in_sizes[0] / (T * C);   // 64
  const int SEG = T / SEGLEN;            // 4 segments -> 2048 waves total

  dim3 grid(SEG, B);                     // 256 blocks
  dim3 block((C / 16) * 32);             // 8 wave32 = 256 threads (one per channel group)
  pcen_wmma_kernel<<<grid, block, 0, stream>>>(x, alpha, delta, root, out, B, T, C);
}